// GNN_86105504350421
// MI455X (gfx1250) — compile-verified
//
#include <hip/hip_runtime.h>
#include <hip/hip_bf16.h>

#define N_NODES 100000
#define IN_DIM  128
#define HID     128
#define OUT_DIM 32

typedef __attribute__((ext_vector_type(2))) float v2f;
typedef __attribute__((ext_vector_type(8))) float v8f;

// ---------------- atomic f32 add (hardware global_atomic_add_f32) ----------
__device__ __forceinline__ void atomAddF32(float* p, float v) {
    unsafeAtomicAdd(p, v);   // no-return HW f32 atomic on CDNA
}

// ---------------- degree / norm ----------------
__global__ void k_zero_cnt(unsigned* __restrict__ cnt) {
    int i = blockIdx.x * blockDim.x + threadIdx.x;
    if (i < N_NODES) cnt[i] = 0u;
}

__global__ void k_count(const long long* __restrict__ dst,
                        unsigned* __restrict__ cnt, int nE) {
    int e = blockIdx.x * blockDim.x + threadIdx.x;
    if (e < nE) atomicAdd(&cnt[(int)dst[e]], 1u);
}

__global__ void k_dinv(const unsigned* __restrict__ cnt,
                       float* __restrict__ dinv) {
    int i = blockIdx.x * blockDim.x + threadIdx.x;
    if (i < N_NODES) dinv[i] = rsqrtf(1.0f + (float)cnt[i]);
}

// ---------------- GEMM1: hs = (x @ W1) * dinv[row]; acc = hs ---------------
// one wave per 16-row tile, full N=128; V_WMMA_F32_16X16X4_F32, K looped by 4
__global__ void __launch_bounds__(128)
k_gemm1(const float* __restrict__ x,      // [N,128]
        const float* __restrict__ W,      // [128,128]
        const float* __restrict__ dinv,
        float* __restrict__ hs,           // buf1
        float* __restrict__ acc)          // buf2
{
    const int wave  = threadIdx.x >> 5;
    const int lane  = threadIdx.x & 31;
    const int mtile = blockIdx.x * 4 + wave;
    if (mtile >= (N_NODES / 16)) return;

    const int m  = lane & 15;       // row within tile (A) / col within tile (B,C)
    const int kh = lane >> 4;       // K-half selector: 0 -> K{0,1}, 1 -> K{2,3}
    const int row = mtile * 16 + m;

    v8f c[8];
    const v8f z = {0.f,0.f,0.f,0.f,0.f,0.f,0.f,0.f};
#pragma unroll
    for (int nt = 0; nt < 8; ++nt) c[nt] = z;

    for (int kk = 0; kk < 128; kk += 4) {
        const int k0 = kk + kh * 2;
        const float2 av = *(const float2*)(x + (size_t)row * 128 + k0);
        v2f a; a.x = av.x; a.y = av.y;
#pragma unroll
        for (int nt = 0; nt < 8; ++nt) {
            const int n = nt * 16 + m;
            v2f b;
            b.x = W[(size_t)k0 * 128 + n];
            b.y = W[(size_t)(k0 + 1) * 128 + n];
            c[nt] = __builtin_amdgcn_wmma_f32_16x16x4_f32(
                        false, a, false, b, (short)0, c[nt], false, false);
        }
    }

    // D layout: VGPR r -> row (r + 8*kh), col = nt*16 + (lane&15)
    float dv[8];
#pragma unroll
    for (int r = 0; r < 8; ++r) dv[r] = dinv[mtile * 16 + kh * 8 + r];
#pragma unroll
    for (int nt = 0; nt < 8; ++nt) {
#pragma unroll
        for (int r = 0; r < 8; ++r) {
            const int orow = mtile * 16 + kh * 8 + r;
            const int ocol = nt * 16 + m;
            const float v = c[nt][r] * dv[r];
            hs [(size_t)orow * 128 + ocol] = v;
            acc[(size_t)orow * 128 + ocol] = v;   // self-loop contribution pre-seeded
        }
    }
}

// ---------------- scatter layer1: acc[d] += hs[s] (128 feats, wave/edge) ----
__global__ void __launch_bounds__(256)
k_scatter1(const long long* __restrict__ src, const long long* __restrict__ dst,
           const float* __restrict__ hs, float* __restrict__ acc, int nE)
{
    const int w = (int)((blockIdx.x * 256 + threadIdx.x) >> 5);
    if (w >= nE) return;
    const int lane = threadIdx.x & 31;
    const int s = (int)src[w];
    const int d = (int)dst[w];
    const float4 v = *(const float4*)(hs + (size_t)s * 128 + lane * 4);
    float* p = acc + (size_t)d * 128 + lane * 4;
    atomAddF32(p + 0, v.x);
    atomAddF32(p + 1, v.y);
    atomAddF32(p + 2, v.z);
    atomAddF32(p + 3, v.w);
}

// ---------------- post1: h = relu(dinv*acc + b1) ---------------------------
__global__ void __launch_bounds__(256)
k_post1(const float* __restrict__ acc, const float* __restrict__ dinv,
        const float* __restrict__ b1, float* __restrict__ h)
{
    const int i = blockIdx.x * 256 + threadIdx.x;
    if (i < N_NODES * HID) {
        const int node = i >> 7;
        const int f    = i & 127;
        const float v = dinv[node] * acc[i] + b1[f];
        h[i] = fmaxf(v, 0.0f);
    }
}

// ---------------- GEMM2: hs2 = (h @ W2) * dinv; acc2 = hs2 -----------------
__global__ void __launch_bounds__(128)
k_gemm2(const float* __restrict__ h,      // [N,128]
        const float* __restrict__ W,      // [128,32]
        const float* __restrict__ dinv,
        float* __restrict__ hs2, float* __restrict__ acc2)
{
    const int wave  = threadIdx.x >> 5;
    const int lane  = threadIdx.x & 31;
    const int mtile = blockIdx.x * 4 + wave;
    if (mtile >= (N_NODES / 16)) return;

    const int m  = lane & 15;
    const int kh = lane >> 4;
    const int row = mtile * 16 + m;

    v8f c[2];
    const v8f z = {0.f,0.f,0.f,0.f,0.f,0.f,0.f,0.f};
    c[0] = z; c[1] = z;

    for (int kk = 0; kk < 128; kk += 4) {
        const int k0 = kk + kh * 2;
        const float2 av = *(const float2*)(h + (size_t)row * 128 + k0);
        v2f a; a.x = av.x; a.y = av.y;
#pragma unroll
        for (int nt = 0; nt < 2; ++nt) {
            const int n = nt * 16 + m;
            v2f b;
            b.x = W[(size_t)k0 * 32 + n];
            b.y = W[(size_t)(k0 + 1) * 32 + n];
            c[nt] = __builtin_amdgcn_wmma_f32_16x16x4_f32(
                        false, a, false, b, (short)0, c[nt], false, false);
        }
    }

    float dv[8];
#pragma unroll
    for (int r = 0; r < 8; ++r) dv[r] = dinv[mtile * 16 + kh * 8 + r];
#pragma unroll
    for (int nt = 0; nt < 2; ++nt) {
#pragma unroll
        for (int r = 0; r < 8; ++r) {
            const int orow = mtile * 16 + kh * 8 + r;
            const int ocol = nt * 16 + m;
            const float v = c[nt][r] * dv[r];
            hs2 [(size_t)orow * 32 + ocol] = v;
            acc2[(size_t)orow * 32 + ocol] = v;
        }
    }
}

// ---------------- scatter layer2: acc2[d] += hs2[s] (32 feats, wave/edge) --
__global__ void __launch_bounds__(256)
k_scatter2(const long long* __restrict__ src, const long long* __restrict__ dst,
           const float* __restrict__ hs2, float* __restrict__ acc2, int nE)
{
    const int w = (int)((blockIdx.x * 256 + threadIdx.x) >> 5);
    if (w >= nE) return;
    const int lane = threadIdx.x & 31;
    const int s = (int)src[w];
    const int d = (int)dst[w];
    atomAddF32(acc2 + (size_t)d * 32 + lane, hs2[(size_t)s * 32 + lane]);
}

// ---------------- post2: logits = dinv*acc2 + b2 ; log_softmax -------------
__global__ void __launch_bounds__(256)
k_post2(const float* __restrict__ acc2, const float* __restrict__ dinv,
        const float* __restrict__ b2, float* __restrict__ out)
{
    const int r = (int)((blockIdx.x * 256 + threadIdx.x) >> 5);
    if (r >= N_NODES) return;
    const int lane = threadIdx.x & 31;

    const float v = dinv[r] * acc2[(size_t)r * 32 + lane] + b2[lane];

    float mx = v;
#pragma unroll
    for (int o = 16; o >= 1; o >>= 1) mx = fmaxf(mx, __shfl_xor(mx, o, 32));
    float s = expf(v - mx);
#pragma unroll
    for (int o = 16; o >= 1; o >>= 1) s += __shfl_xor(s, o, 32);
    out[(size_t)r * 32 + lane] = v - mx - logf(s);
}

// ---------------------------------------------------------------------------
extern "C" void kernel_launch(void* const* d_in, const int* in_sizes, int n_in,
                              void* d_out, int out_size, void* d_ws, size_t ws_size,
                              hipStream_t stream)
{
    const float*     x  = (const float*)d_in[0];
    const long long* ei = (const long long*)d_in[1];   // int64 [2, E]
    const float*     W1 = (const float*)d_in[2];
    const float*     b1 = (const float*)d_in[3];
    const float*     W2 = (const float*)d_in[4];
    const float*     b2 = (const float*)d_in[5];
    float*           out = (float*)d_out;

    const int nE = in_sizes[1] / 2;
    const long long* src = ei;
    const long long* dst = ei + nE;

    // workspace layout
    char* ws = (char*)d_ws;
    float*    dinv = (float*)(ws);                               // 400 KB
    unsigned* cnt  = (unsigned*)(ws + (512 * 1024));             // 400 KB
    float*    buf1 = (float*)(ws + (1u << 20));                  // 51.2 MB: hs1, then relu(h1)
    float*    buf2 = (float*)(ws + (1u << 20)
                              + (size_t)N_NODES * HID * 4);      // 51.2 MB: acc1, then hs2+acc2
    float*    hs2  = buf2;
    float*    acc2 = buf2 + (size_t)N_NODES * OUT_DIM;

    const int MT = N_NODES / 16;   // 6250 row tiles (exact)

    k_zero_cnt<<<(N_NODES + 255) / 256, 256, 0, stream>>>(cnt);
    k_count   <<<(nE + 255) / 256,      256, 0, stream>>>(dst, cnt, nE);
    k_dinv    <<<(N_NODES + 255) / 256, 256, 0, stream>>>(cnt, dinv);

    // layer 1
    k_gemm1   <<<(MT + 3) / 4, 128, 0, stream>>>(x, W1, dinv, buf1, buf2);
    k_scatter1<<<(nE + 7) / 8, 256, 0, stream>>>(src, dst, buf1, buf2, nE);
    k_post1   <<<(N_NODES * HID + 255) / 256, 256, 0, stream>>>(buf2, dinv, b1, buf1);

    // layer 2
    k_gemm2   <<<(MT + 3) / 4, 128, 0, stream>>>(buf1, W2, dinv, hs2, acc2);
    k_scatter2<<<(nE + 7) / 8, 256, 0, stream>>>(src, dst, hs2, acc2, nE);
    k_post2   <<<(N_NODES + 7) / 8, 256, 0, stream>>>(acc2, dinv, b2, out);
}